// GAT_61168924230419
// MI455X (gfx1250) — compile-verified
//
#include <hip/hip_runtime.h>
#include <math.h>

typedef __attribute__((ext_vector_type(2))) float v2f;
typedef __attribute__((ext_vector_type(8))) float v8f;

#define GAT_N     50000
#define GAT_E     800000
#define GAT_FIN   128
#define GAT_HID0  64
#define GAT_HID1  32
#define GAT_HEADS 2
#define NEG_SLOPE 0.2f

// ---------------------------------------------------------------------------
// fp32 WMMA GEMM: D[M,Nc] = A[M,K] @ B[K,Nc].  One wave per 16x16 tile.
// A-frag (16x4 f32): lanes 0-15 -> M=lane, VGPR0/1 = K,K+1 ; lanes 16-31 -> K+2,K+3
// B-frag (4x16 f32): same striping with N in place of M.
// C/D (16x16 f32, 8 VGPRs): lane&15 = N, VGPR r -> M=r (lanes 0-15) / M=r+8 (16-31)
// ---------------------------------------------------------------------------
__global__ void gemm_wmma_f32(const float* __restrict__ A,
                              const float* __restrict__ B,
                              float* __restrict__ D,
                              int M, int Nc, int K) {
    const int tile_m = blockIdx.x * 16;
    const int tile_n = blockIdx.y * 16;
    const int lane   = threadIdx.x & 31;
    const int half   = lane >> 4;     // 0 or 1
    const int l15    = lane & 15;

    v8f acc = {};
    const float* __restrict__ arow = A + (size_t)(tile_m + l15) * K;
    for (int k = 0; k < K; k += 4) {
        const int kk = k + (half << 1);
        v2f a, b;
        a.x = arow[kk];
        a.y = arow[kk + 1];
        b.x = B[(size_t)kk       * Nc + tile_n + l15];
        b.y = B[(size_t)(kk + 1) * Nc + tile_n + l15];
        acc = __builtin_amdgcn_wmma_f32_16x16x4_f32(
            /*neg_a=*/false, a, /*neg_b=*/false, b,
            /*c_mod=*/(short)0, acc, /*reuse_a=*/false, /*reuse_b=*/false);
    }
    const int col = tile_n + l15;
    const int row0 = tile_m + half * 8;
#pragma unroll
    for (int r = 0; r < 8; ++r)
        D[(size_t)(row0 + r) * Nc + col] = acc[r];
}

// ---------------------------------------------------------------------------
// per-node attention coefficients: asrc[n,h] = <H[n,h,:], att_src[h,:]>
// ---------------------------------------------------------------------------
__global__ void attn_coef(const float* __restrict__ H,
                          const float* __restrict__ att_src,
                          const float* __restrict__ att_dst,
                          float* __restrict__ asrc,
                          float* __restrict__ adst,
                          int n, int C) {
    const int i = blockIdx.x * blockDim.x + threadIdx.x;
    if (i >= n) return;
#pragma unroll
    for (int h = 0; h < GAT_HEADS; ++h) {
        const float* row = H + (size_t)i * GAT_HEADS * C + (size_t)h * C;
        float s = 0.f, d = 0.f;
        for (int c = 0; c < C; ++c) {
            const float v = row[c];
            s += v * att_src[h * C + c];
            d += v * att_dst[h * C + c];
        }
        asrc[i * GAT_HEADS + h] = s;
        adst[i * GAT_HEADS + h] = d;
    }
}

__global__ void fill_f32(float* __restrict__ p, float v, int n) {
    const int i = blockIdx.x * blockDim.x + threadIdx.x;
    if (i < n) p[i] = v;
}

// order-independent (deterministic) float atomic max, init must be -inf
__device__ __forceinline__ void atomicMaxFloat(float* addr, float value) {
    if (value >= 0.0f) atomicMax((int*)addr, __float_as_int(value));
    else               atomicMin((unsigned int*)addr, __float_as_uint(value));
}

__device__ __forceinline__ void edge_endpoints(const int* __restrict__ ei,
                                               int e, int E_, int& src, int& dst) {
    if (e < E_) { src = ei[e]; dst = ei[E_ + e]; }
    else        { src = dst = e - E_; }           // appended self-loops
}

// pass A: segment-max of leaky_relu(a_src[src]+a_dst[dst]) per dst,head
__global__ void edge_max(const int* __restrict__ ei, int E_, int n,
                         const float* __restrict__ asrc,
                         const float* __restrict__ adst,
                         float* __restrict__ m) {
    const int e = blockIdx.x * blockDim.x + threadIdx.x;
    if (e >= E_ + n) return;
    int src, dst;
    edge_endpoints(ei, e, E_, src, dst);
#pragma unroll
    for (int h = 0; h < GAT_HEADS; ++h) {
        float v = asrc[src * GAT_HEADS + h] + adst[dst * GAT_HEADS + h];
        v = v > 0.f ? v : NEG_SLOPE * v;
        atomicMaxFloat(&m[dst * GAT_HEADS + h], v);
    }
}

// pass B: p = exp(e - m[dst]); store p, segment-sum into s[dst]
__global__ void edge_exp_sum(const int* __restrict__ ei, int E_, int n,
                             const float* __restrict__ asrc,
                             const float* __restrict__ adst,
                             const float* __restrict__ m,
                             float* __restrict__ pbuf,
                             float* __restrict__ s) {
    const int e = blockIdx.x * blockDim.x + threadIdx.x;
    if (e >= E_ + n) return;
    int src, dst;
    edge_endpoints(ei, e, E_, src, dst);
#pragma unroll
    for (int h = 0; h < GAT_HEADS; ++h) {
        float v = asrc[src * GAT_HEADS + h] + adst[dst * GAT_HEADS + h];
        v = v > 0.f ? v : NEG_SLOPE * v;
        const float p = expf(v - m[dst * GAT_HEADS + h]);
        pbuf[(size_t)e * GAT_HEADS + h] = p;
        atomicAdd(&s[dst * GAT_HEADS + h], p);
    }
}

// pass C: agg[dst, h, c] += H[src, h, c] * p/(s[dst,h]+eps)
// one thread per (edge, channel) -> fully coalesced gather & atomic scatter
__global__ void edge_aggregate(const int* __restrict__ ei, int E_, int n,
                               const float* __restrict__ H,
                               const float* __restrict__ pbuf,
                               const float* __restrict__ s,
                               float* __restrict__ agg, int C) {
    const int HC = GAT_HEADS * C;
    const long long t = (long long)blockIdx.x * blockDim.x + threadIdx.x;
    const long long total = (long long)(E_ + n) * HC;
    if (t >= total) return;
    const int e = (int)(t / HC);
    const int j = (int)(t % HC);
    const int h = j / C;
    int src, dst;
    edge_endpoints(ei, e, E_, src, dst);
    const float alpha = pbuf[(size_t)e * GAT_HEADS + h] /
                        (s[dst * GAT_HEADS + h] + 1e-16f);
    atomicAdd(&agg[(size_t)dst * HC + j], H[(size_t)src * HC + j] * alpha);
}

__global__ void bias_relu_inplace(float* __restrict__ a,
                                  const float* __restrict__ bias,
                                  int total, int HC) {
    const int i = blockIdx.x * blockDim.x + threadIdx.x;
    if (i >= total) return;
    const float v = a[i] + bias[i % HC];
    a[i] = v > 0.f ? v : 0.f;
}

// mean over heads + bias2
__global__ void finalize_mean(const float* __restrict__ agg,
                              const float* __restrict__ bias,
                              float* __restrict__ out, int n) {
    const int i = blockIdx.x * blockDim.x + threadIdx.x;
    if (i >= n * GAT_HID1) return;
    const int node = i / GAT_HID1;
    const int c    = i % GAT_HID1;
    const float a0 = agg[(size_t)node * GAT_HEADS * GAT_HID1 + c];
    const float a1 = agg[(size_t)node * GAT_HEADS * GAT_HID1 + GAT_HID1 + c];
    out[i] = 0.5f * (a0 + a1) + bias[c];
}

// ---------------------------------------------------------------------------
extern "C" void kernel_launch(void* const* d_in, const int* in_sizes, int n_in,
                              void* d_out, int out_size, void* d_ws, size_t ws_size,
                              hipStream_t stream) {
    (void)in_sizes; (void)n_in; (void)out_size; (void)ws_size;
    const float* x        = (const float*)d_in[0];
    const int*   ei       = (const int*)  d_in[1];   // [2,E] int32 per harness
    const float* W1       = (const float*)d_in[2];
    const float* att_src1 = (const float*)d_in[3];
    const float* att_dst1 = (const float*)d_in[4];
    const float* bias1    = (const float*)d_in[5];
    const float* W2       = (const float*)d_in[6];
    const float* att_src2 = (const float*)d_in[7];
    const float* att_dst2 = (const float*)d_in[8];
    const float* bias2    = (const float*)d_in[9];
    float* out = (float*)d_out;

    // workspace layout (floats)
    float* ws   = (float*)d_ws;
    float* h1   = ws;                    // N*128 = 6.40M
    float* agg1 = ws + 6400000;          // N*128 = 6.40M  (becomes h1r in place)
    float* asrc = ws + 12800000;         // N*2
    float* adst = ws + 12900000;         // N*2
    float* mbuf = ws + 13000000;         // N*2
    float* sbuf = ws + 13100000;         // N*2
    float* pbuf = ws + 13200000;         // (E+N)*2 = 1.70M
    float* h2   = h1;                    // N*64 = 3.20M (reuse)
    float* agg2 = h1 + 3200000;          // N*64 = 3.20M (reuse)

    const int Etot = GAT_E + GAT_N;      // appended self-loops
    const int TB = 256;
    const float NEG_INF = -__builtin_huge_valf();

    // ---------------- layer 1 ----------------
    {
        dim3 grid(GAT_N / 16, (GAT_HEADS * GAT_HID0) / 16);
        gemm_wmma_f32<<<grid, 32, 0, stream>>>(x, W1, h1,
                                               GAT_N, GAT_HEADS * GAT_HID0, GAT_FIN);
    }
    attn_coef<<<(GAT_N + 127) / 128, 128, 0, stream>>>(h1, att_src1, att_dst1,
                                                       asrc, adst, GAT_N, GAT_HID0);
    fill_f32<<<(GAT_N * 2 + TB - 1) / TB, TB, 0, stream>>>(mbuf, NEG_INF, GAT_N * 2);
    fill_f32<<<(GAT_N * 2 + TB - 1) / TB, TB, 0, stream>>>(sbuf, 0.f, GAT_N * 2);
    fill_f32<<<(GAT_N * 128 + TB - 1) / TB, TB, 0, stream>>>(agg1, 0.f, GAT_N * 128);

    edge_max<<<(Etot + TB - 1) / TB, TB, 0, stream>>>(ei, GAT_E, GAT_N, asrc, adst, mbuf);
    edge_exp_sum<<<(Etot + TB - 1) / TB, TB, 0, stream>>>(ei, GAT_E, GAT_N, asrc, adst,
                                                          mbuf, pbuf, sbuf);
    {
        const long long total = (long long)Etot * (GAT_HEADS * GAT_HID0);
        edge_aggregate<<<(unsigned)((total + TB - 1) / TB), TB, 0, stream>>>(
            ei, GAT_E, GAT_N, h1, pbuf, sbuf, agg1, GAT_HID0);
    }
    bias_relu_inplace<<<(GAT_N * 128 + TB - 1) / TB, TB, 0, stream>>>(
        agg1, bias1, GAT_N * 128, GAT_HEADS * GAT_HID0);

    // ---------------- layer 2 ----------------
    {
        dim3 grid(GAT_N / 16, (GAT_HEADS * GAT_HID1) / 16);
        gemm_wmma_f32<<<grid, 32, 0, stream>>>(agg1, W2, h2,
                                               GAT_N, GAT_HEADS * GAT_HID1,
                                               GAT_HEADS * GAT_HID0);
    }
    attn_coef<<<(GAT_N + 127) / 128, 128, 0, stream>>>(h2, att_src2, att_dst2,
                                                       asrc, adst, GAT_N, GAT_HID1);
    fill_f32<<<(GAT_N * 2 + TB - 1) / TB, TB, 0, stream>>>(mbuf, NEG_INF, GAT_N * 2);
    fill_f32<<<(GAT_N * 2 + TB - 1) / TB, TB, 0, stream>>>(sbuf, 0.f, GAT_N * 2);
    fill_f32<<<(GAT_N * 64 + TB - 1) / TB, TB, 0, stream>>>(agg2, 0.f, GAT_N * 64);

    edge_max<<<(Etot + TB - 1) / TB, TB, 0, stream>>>(ei, GAT_E, GAT_N, asrc, adst, mbuf);
    edge_exp_sum<<<(Etot + TB - 1) / TB, TB, 0, stream>>>(ei, GAT_E, GAT_N, asrc, adst,
                                                          mbuf, pbuf, sbuf);
    {
        const long long total = (long long)Etot * (GAT_HEADS * GAT_HID1);
        edge_aggregate<<<(unsigned)((total + TB - 1) / TB), TB, 0, stream>>>(
            ei, GAT_E, GAT_N, h2, pbuf, sbuf, agg2, GAT_HID1);
    }
    finalize_mean<<<(GAT_N * GAT_HID1 + TB - 1) / TB, TB, 0, stream>>>(
        agg2, bias2, out, GAT_N);
}